// CRsAE1D_5918464934080
// MI455X (gfx1250) — compile-verified
//
#include <hip/hip_runtime.h>
#include <cmath>

// Problem constants (match reference)
#define BQ   32
#define NN   8192
#define CC   64
#define KK   31
#define MM   (NN - KK + 1)   // 8162
#define TT   30
#define INVL 0.2f            // 1/L, L=5
#define THR  0.02f           // LAM/L = 0.1/5

typedef float v2f __attribute__((ext_vector_type(2)));
typedef float v8f __attribute__((ext_vector_type(8)));

#if defined(__HIP_DEVICE_COMPILE__) && __has_builtin(__builtin_amdgcn_global_load_async_to_lds_b32)
#define HAS_ASYNC_LDS 1
#else
#define HAS_ASYNC_LDS 0
#endif

// ---------------------------------------------------------------------------
// Zero-init both x buffers (x_tmp in ws, x_old in d_out x-region).
// ---------------------------------------------------------------------------
__global__ __launch_bounds__(256) void crsae_zero_kernel(float* __restrict__ a,
                                                         float* __restrict__ b,
                                                         long n) {
    long i = (long)blockIdx.x * 256 + threadIdx.x;
    if (i < n) { a[i] = 0.f; b[i] = 0.f; }
}

// ---------------------------------------------------------------------------
// Hop (conv_transpose1d, single output channel) — matrix-vector, VALU kernel.
//   subtract=1:  dst[b,n] = y[b,n] - sum_{c,k} x[b,c,n-k] H[c,k]   (residual)
//   subtract=0:  dst[b,n] =          sum_{c,k} x[b,c,n-k] H[c,k]   (y_hat)
// block = 128 threads, n-tile = 128; x tile (64ch x 158) staged into LDS via
// GLOBAL_LOAD_ASYNC_TO_LDS_B32 (ASYNCcnt), halo zeros via direct DS stores.
// ---------------------------------------------------------------------------
__global__ __launch_bounds__(128) void crsae_hop_kernel(
    const float* __restrict__ x, const float* __restrict__ H,
    const float* __restrict__ y, float* __restrict__ dst, int subtract) {
    __shared__ float xs[CC][160];   // 158 used (128 + 30 halo), padded
    __shared__ float hs[CC][32];    // K padded to 32

    const int b   = blockIdx.y;
    const int n0  = blockIdx.x * 128;
    const int tid = threadIdx.x;

    for (int i = tid; i < CC * 32; i += 128) {
        int c = i >> 5, k = i & 31;
        hs[c][k] = (k < KK) ? H[c * KK + k] : 0.f;
    }
    const float* xb = x + (size_t)b * CC * MM;
    for (int i = tid; i < CC * 158; i += 128) {
        int c = i / 158, mm = i % 158;
        int m = n0 - 30 + mm;
        if (m >= 0 && m < MM) {
#if HAS_ASYNC_LDS
            __builtin_amdgcn_global_load_async_to_lds_b32(
                (__attribute__((address_space(1))) int*)(void*)
                    (xb + (size_t)c * MM + m),
                (__attribute__((address_space(3))) int*)(void*)&xs[c][mm],
                0, 0);
#else
            xs[c][mm] = xb[(size_t)c * MM + m];
#endif
        } else {
            xs[c][mm] = 0.f;
        }
    }
#if HAS_ASYNC_LDS
#if __has_builtin(__builtin_amdgcn_s_wait_asynccnt)
    __builtin_amdgcn_s_wait_asynccnt(0);
#else
    asm volatile("s_wait_asynccnt 0" ::: "memory");
#endif
#endif
    __syncthreads();

    float acc = 0.f;
    for (int c = 0; c < CC; ++c) {
        #pragma unroll
        for (int k = 0; k < KK; ++k)
            acc = fmaf(xs[c][tid + 30 - k], hs[c][k], acc);
    }
    const size_t on = (size_t)b * NN + n0 + tid;
    dst[on] = subtract ? (y[on] - acc) : acc;
}

// ---------------------------------------------------------------------------
// HT (VALID conv) as fp32 WMMA GEMM + fused FISTA update.
//   dot[c,m] = sum_k res[b, m+k] * H[c,k]
//   x_new    = relu(x_tmp + dot/L - thr)
//   x_old   <- x_new ;  x_tmp <- x_new + mu*(x_new - x_old_prev)
// D(16c x 16m) = A(16c x 4k = H) x B(4k x 16m = res window), 8 k-chunks.
// block = 256 threads = 8 waves; tile = 64c x 64m (16 WMMA tiles, 2/wave).
// ---------------------------------------------------------------------------
__global__ __launch_bounds__(256) void crsae_ht_update_kernel(
    const float* __restrict__ res, const float* __restrict__ H,
    float* __restrict__ x_old, float* __restrict__ x_tmp, float mu) {
    __shared__ float rs[128];       // res[m0 .. m0+93] (zero-padded)
    __shared__ float hsk[CC][32];   // H[c][k], K padded to 32

    const int b   = blockIdx.y;
    const int m0  = blockIdx.x * 64;
    const int tid = threadIdx.x;

    for (int i = tid; i < CC * 32; i += 256) {
        int c = i >> 5, k = i & 31;
        hsk[c][k] = (k < KK) ? H[c * KK + k] : 0.f;
    }
    if (tid < 128) {
        int n = m0 + tid;
        rs[tid] = (tid < MM - m0 + KK - 1 && n < NN) ? res[(size_t)b * NN + n] : 0.f;
    }
    __syncthreads();

    const int wave = tid >> 5;
    const int lane = tid & 31;
    const int half = lane >> 4;     // 0: K{0,1}, 1: K{2,3} within a chunk
    const int l16  = lane & 15;
    const int c0   = (wave >> 1) * 16;

    #pragma unroll
    for (int sub = 0; sub < 2; ++sub) {
        const int mt    = (wave & 1) * 2 + sub;
        const int mbase = mt * 16;
        v8f acc = {0.f, 0.f, 0.f, 0.f, 0.f, 0.f, 0.f, 0.f};

        #pragma unroll
        for (int kk = 0; kk < 8; ++kk) {
            const int k0 = kk * 4 + half * 2;
            v2f a, bb;
            // A (16x4): row = c (lanes hold rows 0-15 twice), K split by half
            a.x  = hsk[c0 + l16][k0];
            a.y  = hsk[c0 + l16][k0 + 1];
            // B (4x16): col = m (lane 0-15), K row split by half
            bb.x = rs[mbase + l16 + k0];
            bb.y = rs[mbase + l16 + k0 + 1];
            acc = __builtin_amdgcn_wmma_f32_16x16x4_f32(
                false, a, false, bb, (short)0, acc, false, false);
        }

        // D layout: VGPR v, lanes 0-15 -> row v, lanes 16-31 -> row v+8; col = l16
        const int m = m0 + mbase + l16;
        if (m < MM) {
            #pragma unroll
            for (int v = 0; v < 8; ++v) {
                const int c  = c0 + v + 8 * half;
                const size_t idx = ((size_t)b * CC + c) * MM + m;
                const float xo = x_old[idx];
                const float xt = x_tmp[idx];
                float xn = fmaf(acc[v], INVL, xt) - THR;
                xn = xn > 0.f ? xn : 0.f;
                x_old[idx] = xn;                      // x_new (next iter's x_old)
                x_tmp[idx] = fmaf(mu, xn - xo, xn);   // next iter's x_tmp
            }
        }
    }
}

// ---------------------------------------------------------------------------
extern "C" void kernel_launch(void* const* d_in, const int* in_sizes, int n_in,
                              void* d_out, int out_size, void* d_ws, size_t ws_size,
                              hipStream_t stream) {
    const float* y = (const float*)d_in[0];   // [B,1,N]
    const float* H = (const float*)d_in[1];   // [C,1,K]

    float* out   = (float*)d_out;
    float* yhat  = out;                              // B*N
    float* x_old = out + (size_t)BQ * NN;            // B*C*M (final x_new lands here)

    float* ws    = (float*)d_ws;
    float* res   = ws;                               // B*N
    float* x_tmp = ws + (size_t)BQ * NN;             // B*C*M

    const long nx = (long)BQ * CC * MM;
    crsae_zero_kernel<<<dim3((unsigned)((nx + 255) / 256)), 256, 0, stream>>>(
        x_tmp, x_old, nx);

    const dim3 hop_grid(NN / 128, BQ);
    const dim3 ht_grid((MM + 63) / 64, BQ);

    double s = 1.0;
    for (int t = 0; t < TT; ++t) {
        // res = y - Hop(x_tmp)
        crsae_hop_kernel<<<hop_grid, 128, 0, stream>>>(x_tmp, H, y, res, 1);
        // momentum scalar (host-side, deterministic)
        const double snew = (1.0 + sqrt(1.0 + 4.0 * s * s)) * 0.5;
        const float  mu   = (float)((s - 1.0) / snew);
        s = snew;
        // x_new = relu(x_tmp + HT(res)/L - thr); momentum update (WMMA kernel)
        crsae_ht_update_kernel<<<ht_grid, 256, 0, stream>>>(res, H, x_old, x_tmp, mu);
    }
    // y_hat = Hop(x_new)   (x_new already in d_out x-region)
    crsae_hop_kernel<<<hop_grid, 128, 0, stream>>>(x_old, H, y, yhat, 0);
}